// GCN_Autoencoder_75866302317043
// MI455X (gfx1250) — compile-verified
//
#include <hip/hip_runtime.h>
#include <hip/hip_bf16.h>

// GCN autoencoder for MI455X (gfx1250, wave32).
//  - fp32 WMMA (v_wmma_f32_16x16x4_f32) for the dense h = x@W GEMMs
//  - L2-resident edge scatter with global_atomic_add_f32
// Precision kept at fp32 to match the reference bit-for-bit semantics
// (GEMM cost is negligible; scatter dominates and is L2-bound on 192MB L2).

typedef float v2f __attribute__((ext_vector_type(2)));
typedef float v8f __attribute__((ext_vector_type(8)));

#define FEAT 128

// XOR swizzle for LDS-resident transposed W: sW[swz(n,k)] = W[k][n].
// XOR value has bit0 == 0 so b64 (float2) pairs stay contiguous; bits 2..5
// spread the 16 column-lanes across all 64 LDS banks -> conflict-free B reads.
__device__ __forceinline__ int swz(int n, int k) {
  return (n << 7) + (k ^ ((n & 15) << 2));
}

// ---------------------------------------------------------------------------
// GEMM: Y[nrows x 128] = (relu?)(X[nrows x 128]) @ W[128 x 128]
// 256 threads = 8 wave32; each wave owns a 16-row strip, all 128 columns.
// Per wave: 32 K-steps x 8 N-tiles of V_WMMA_F32_16X16X4_F32.
// ---------------------------------------------------------------------------
__global__ __launch_bounds__(256) void k_gemm_wmma(
    const float* __restrict__ X, const float* __restrict__ Wg,
    float* __restrict__ Y, int nrows, int relu_in)
{
  __shared__ float sW[FEAT * FEAT];  // exactly 64 KB

  const int t = threadIdx.x;
  const int row0 = blockIdx.x * 128;

  // Stage W transposed+swizzled: coalesced global reads.
#pragma unroll
  for (int i = 0; i < 64; ++i) {
    int idx = i * 256 + t;            // 16384 elements
    int k = idx >> 7;
    int n = idx & 127;
    sW[swz(n, k)] = Wg[idx];
  }
  __syncthreads();

  const int wave = t >> 5;
  const int lane = t & 31;
  const int l16 = lane & 15;
  const int hi = lane >> 4;          // 0: K=0,1  1: K=2,3  (ISA A-frag layout)

  int arow = row0 + wave * 16 + l16;
  if (arow >= nrows) arow = nrows - 1;      // clamp reads; stores are guarded
  const float* __restrict__ xr = X + (size_t)arow * FEAT + 2 * hi;

  v8f zero = {};
  v8f acc[8];
#pragma unroll
  for (int j = 0; j < 8; ++j) acc[j] = zero;

  for (int k0 = 0; k0 < FEAT; k0 += 4) {
    // A-frag (16x4 f32): lanes 0-15 hold X[row][k0..k0+1], lanes 16-31 k0+2..3
    v2f a = *(const v2f*)(xr + k0);
    if (relu_in) { a.x = fmaxf(a.x, 0.0f); a.y = fmaxf(a.y, 0.0f); }
    const int kk = k0 + 2 * hi;
#pragma unroll
    for (int j = 0; j < 8; ++j) {
      // B-frag (4x16 f32): lane holds column n = j*16+l16, K pair per `hi`
      v2f b = *(const v2f*)&sW[swz(j * 16 + l16, kk)];
      acc[j] = __builtin_amdgcn_wmma_f32_16x16x4_f32(
          /*neg_a=*/false, a, /*neg_b=*/false, b,
          /*c_mod=*/(short)0, acc[j], /*reuse_a=*/false, /*reuse_b=*/false);
    }
  }

  // C/D layout: VGPR v -> M = 8*hi + v, N = l16 (per tile j).
#pragma unroll
  for (int j = 0; j < 8; ++j) {
    const int col = j * 16 + l16;
#pragma unroll
    for (int v = 0; v < 8; ++v) {
      const int gr = row0 + wave * 16 + 8 * hi + v;
      if (gr < nrows) Y[(size_t)gr * FEAT + col] = acc[j][v];
    }
  }
}

// ---------------------------------------------------------------------------
// Degree / normalization helpers
// ---------------------------------------------------------------------------
__global__ void k_fill1(float* __restrict__ d, int n) {
  int i = blockIdx.x * blockDim.x + threadIdx.x;
  if (i < n) d[i] = 1.0f;               // self-loop contributes 1 to every node
}

__global__ void k_deg(const int* __restrict__ dst, float* __restrict__ deg, int E) {
  int e = blockIdx.x * blockDim.x + threadIdx.x;
  if (e < E) atomicAdd(&deg[dst[e]], 1.0f);
}

__global__ void k_rsqrt(float* __restrict__ d, int n) {
  int i = blockIdx.x * blockDim.x + threadIdx.x;
  if (i < n) {
    float v = d[i];
    d[i] = (v > 0.0f) ? rsqrtf(v) : 0.0f;
  }
}

// dest = bias + dinv^2 * hW   (fuses the self-loop edge; fully overwrites dest)
__global__ void k_init_dest(const float* __restrict__ hW,
                            const float* __restrict__ dinv,
                            const float* __restrict__ bias,
                            float* __restrict__ dest, int nf) {
  int i = blockIdx.x * blockDim.x + threadIdx.x;
  if (i < nf) {
    int node = i >> 7;
    int c = i & 127;
    float di = dinv[node];
    dest[i] = bias[c] + di * di * hW[i];
  }
}

// Edge scatter: one thread per (edge, 4 columns) -> float4 gather + 4 atomics.
__global__ void k_scatter(const int* __restrict__ src, const int* __restrict__ dst,
                          const float* __restrict__ dinv, const float* __restrict__ hW,
                          float* __restrict__ dest, int E) {
  int idx = blockIdx.x * blockDim.x + threadIdx.x;
  if (idx >= E * 32) return;
  int e = idx >> 5;
  int c4 = (idx & 31) << 2;
  int s = src[e];
  int d = dst[e];
  float w = dinv[s] * dinv[d];
  const float4 v = *(const float4*)(hW + ((size_t)s << 7) + c4);
  float* o = dest + ((size_t)d << 7) + c4;
  atomicAdd(o + 0, v.x * w);
  atomicAdd(o + 1, v.y * w);
  atomicAdd(o + 2, v.z * w);
  atomicAdd(o + 3, v.w * w);
}

__global__ void k_relu(float* __restrict__ d, int n) {
  int i = blockIdx.x * blockDim.x + threadIdx.x;
  if (i < n) d[i] = fmaxf(d[i], 0.0f);
}

// ---------------------------------------------------------------------------
// Launch
// ---------------------------------------------------------------------------
extern "C" void kernel_launch(void* const* d_in, const int* in_sizes, int n_in,
                              void* d_out, int out_size, void* d_ws, size_t ws_size,
                              hipStream_t stream) {
  const float* x  = (const float*)d_in[0];
  const int*   ei = (const int*)d_in[1];          // edge_index [2, E] (int32)
  const float* Wm[4] = {(const float*)d_in[2], (const float*)d_in[4],
                        (const float*)d_in[6], (const float*)d_in[8]};
  const float* Bm[4] = {(const float*)d_in[3], (const float*)d_in[5],
                        (const float*)d_in[7], (const float*)d_in[9]};

  const int N = in_sizes[0] / FEAT;               // 50000
  const int E = in_sizes[1] / 2;                  // 800000
  const int* src = ei;
  const int* dstv = ei + E;

  float* dinv = (float*)d_ws;                     // N floats (deg -> dinv in place)
  float* H1   = dinv + ((N + 127) & ~127);        // hW scratch, N*128 floats
  float* H2   = H1 + (size_t)N * FEAT;            // feature ping-pong buffer
  float* outp = (float*)d_out;

  const int nf = N * FEAT;

  // GCN norm: deg = 1 (self-loop) + in-degree; dinv = rsqrt(deg)
  k_fill1<<<(N + 255) / 256, 256, 0, stream>>>(dinv, N);
  k_deg<<<(E + 255) / 256, 256, 0, stream>>>(dstv, dinv, E);
  k_rsqrt<<<(N + 255) / 256, 256, 0, stream>>>(dinv, N);

  // 4 GCN layers: buffers rotate  x -> H2 -> d_out -> H2 -> d_out
  const float* in = x;
  for (int l = 0; l < 4; ++l) {
    float* dest = (l & 1) ? outp : H2;
    k_gemm_wmma<<<(N + 127) / 128, 256, 0, stream>>>(in, Wm[l], H1, N, l ? 1 : 0);
    k_init_dest<<<(nf + 255) / 256, 256, 0, stream>>>(H1, dinv, Bm[l], dest, nf);
    k_scatter<<<(E * 32 + 255) / 256, 256, 0, stream>>>(src, dstv, dinv, H1, dest, E);
    in = dest;
  }

  // ReLU of layers 1..3 is fused into the next GEMM's A-load; final layer here.
  k_relu<<<(nf + 255) / 256, 256, 0, stream>>>(outp, nf);
}